// SOMISSMLayer_33200097198711
// MI455X (gfx1250) — compile-verified
//
#include <hip/hip_runtime.h>
#include <hip/hip_bf16.h>
#include <math.h>

// ---------------------------------------------------------------------------
// SOMISSM settle layer for MI455X (gfx1250), f32 WMMA (V_WMMA_F32_16X16X4_F32)
// B operands pre-packed into native WMMA lane layout; M=32 token tiles.
// ---------------------------------------------------------------------------

typedef __attribute__((ext_vector_type(2))) float v2f;
typedef __attribute__((ext_vector_type(8))) float v8f;

constexpr int   HD     = 1024;      // hidden width
constexpr int   BNK    = 64;        // gate bottleneck
constexpr int   NTOK   = 4 * 2048;  // BATCH * SEQ
constexpr int   NSET   = 5;         // int(pi / (sqrt(3)*0.35)) clamped [3,10]
constexpr float DT_    = 0.35f;
constexpr float BETA_  = 2.4248711306f;  // 2*0.7*sqrt(3)
constexpr int   PST    = HD + 4;    // phi LDS row stride: 4112B -> 4-bank row step
constexpr int   HST    = BNK + 4;   // hid LDS row stride: 272B  -> 4-bank row step

__device__ __forceinline__ v8f wmma_f32(v2f a, v2f b, v8f c) {
    // D = A(16x4 f32) * B(4x16 f32) + C(16x16 f32)
    return __builtin_amdgcn_wmma_f32_16x16x4_f32(
        /*neg_a=*/false, a, /*neg_b=*/false, b,
        /*c_mod=*/(short)0, c, /*reuse_a=*/false, /*reuse_b=*/false);
}

// ------------------------- prep kernels ------------------------------------

// deg[o] = max(sum_i |W[o,i]|, 1e-8); im[o] = 1/mass[o]
__global__ __launch_bounds__(256) void k_deg(const float* __restrict__ W,
                                             const float* __restrict__ mass,
                                             float* __restrict__ deg,
                                             float* __restrict__ im) {
    const int o = blockIdx.x;
    const int tid = threadIdx.x;
    float s = 0.f;
    for (int i = tid; i < HD; i += 256) s += fabsf(W[(size_t)o * HD + i]);
    for (int off = 16; off; off >>= 1) s += __shfl_down(s, off);
    __shared__ float rs[8];
    if ((tid & 31) == 0) rs[tid >> 5] = s;
    __syncthreads();
    if (tid == 0) {
        float a = 0.f;
        for (int i = 0; i < 8; ++i) a += rs[i];
        deg[o] = fmaxf(a, 1e-8f);
        im[o]  = 1.f / mass[o];
    }
}

// Pack L_rw^T into WMMA-B lane layout: Lp[ct][ks][lane][j]
//   value = L_rw[o, i] = (i==o) - |W[o,i]|/deg[o]
//   i (K idx) = ks*4 + 2*(lane>>4) + j,  o (N idx) = ct*16 + (lane&15)
__global__ __launch_bounds__(256) void k_ltp(const float* __restrict__ W,
                                             const float* __restrict__ deg,
                                             float* __restrict__ Lp) {
    const int t  = blockIdx.x * 256 + threadIdx.x;   // over HD*HD
    const int j  = t & 1;
    const int L  = (t >> 1) & 31;
    const int ks = (t >> 6) & 255;
    const int ct = t >> 14;
    const int i  = ks * 4 + ((L >> 4) << 1) + j;
    const int o  = ct * 16 + (L & 15);
    Lp[t] = (i == o ? 1.f : 0.f) - fabsf(W[(size_t)o * HD + i]) / deg[o];
}

// Pack gate weights into WMMA-B lane layout.
//   g1p[ct(4)][ks(256)][lane][j] = gate_w1[n, k]  (k = K over HD, n over BNK)
//   g2p[ct(64)][ks(16)][lane][j] = gate_w2[n, k]  (k = K over BNK, n over HD)
__global__ __launch_bounds__(256) void k_gp(const float* __restrict__ g1,
                                            const float* __restrict__ g2,
                                            float* __restrict__ g1p,
                                            float* __restrict__ g2p) {
    const int t  = blockIdx.x * 256 + threadIdx.x;   // over HD*BNK = 65536
    const int j  = t & 1;
    const int L  = (t >> 1) & 31;
    const int kg = ((L >> 4) << 1) + j;
    const int lm = L & 15;
    { const int ks = (t >> 6) & 255, ct = t >> 14;                 // ct 0..3
      g1p[t] = g1[(size_t)(ct * 16 + lm) * HD + ks * 4 + kg]; }
    { const int ks = (t >> 6) & 15,  ct = t >> 10;                 // ct 0..63
      g2p[t] = g2[(size_t)(ct * 16 + lm) * BNK + ks * 4 + kg]; }
}

__global__ __launch_bounds__(256) void k_init(const float* __restrict__ h,
                                              float* __restrict__ phi,
                                              float* __restrict__ pd) {
    const size_t t = (size_t)blockIdx.x * 256 + threadIdx.x;
    phi[t] = h[t];
    pd[t]  = 0.f;
}

// ------------------------- fused settle step -------------------------------
// One block = 32 tokens, 8 wave32. All waves build the 32x64 hidden tile.
// Then each wave owns 128 output columns with TWO row-tile accumulators, so
// every packed-B global_load_b64 feeds two WMMAs.
__global__ __launch_bounds__(256) void k_step(
    const float* __restrict__ hT,  const float* __restrict__ Lp,
    const float* __restrict__ g1p, const float* __restrict__ gb1,
    const float* __restrict__ g2p, const float* __restrict__ gb2,
    const float* __restrict__ prec, const float* __restrict__ im,
    float* __restrict__ phi,        float* __restrict__ phidot) {
    __shared__ __align__(16) float phiS[32 * PST];   // ~128.5 KB
    __shared__ __align__(16) float hidS[32 * HST];   // ~8.5 KB

    const int tid  = threadIdx.x;
    const int wid  = tid >> 5;
    const int lane = tid & 31;
    const int t0   = blockIdx.x * 32;

    // ---- stage 32x1024 phi tile into padded LDS (float4 traffic) ----
    {
        const float4* src = reinterpret_cast<const float4*>(phi + (size_t)t0 * HD);
        for (int i = tid; i < 32 * (HD / 4); i += 256) {
            const int row = i >> 8;          // HD/4 = 256 float4 per row
            const int c4  = i & 255;
            const float4 v = src[row * (HD / 4) + c4];
            float* d = &phiS[row * PST + c4 * 4];
            d[0] = v.x; d[1] = v.y; d[2] = v.z; d[3] = v.w;
        }
    }
    __syncthreads();

    const int lm  = lane & 15;           // A row / B column within 16-tile
    const int kg  = (lane >> 4) << 1;    // k sub-offset: lanes 16-31 hold K+2,K+3
    const int mhi = (lane >> 4) << 3;    // C rows: lanes 16-31 hold M+8

    // ---- hidden tile: hid = relu(phi @ gw1^T + b1), 8 tiles across 8 waves --
    {
        const int mt = wid >> 2;                   // row tile 0/1
        const int n0 = (wid & 3) * 16;             // bottleneck column tile
        v8f acc = {};
        const float* Ab = phiS + (mt * 16 + lm) * PST + kg;
        const v2f*   Bb = reinterpret_cast<const v2f*>(g1p + (size_t)(wid & 3) * 256 * 64) + lane;
        #pragma unroll 4
        for (int ks = 0; ks < 256; ++ks) {
            v2f a = { Ab[ks * 4], Ab[ks * 4 + 1] };
            acc = wmma_f32(a, Bb[ks * 32], acc);
        }
        const float bias = gb1[n0 + lm];
        #pragma unroll
        for (int r = 0; r < 8; ++r) {
            const float v = acc[r] + bias;
            hidS[(mt * 16 + r + mhi) * HST + n0 + lm] = v > 0.f ? v : 0.f;
        }
    }
    __syncthreads();

    // ---- per-wave: 8 column tiles of 16, two row tiles each ----
    for (int t = 0; t < 8; ++t) {
        const int n0  = wid * 128 + t * 16;
        const int col = n0 + lm;

        // gate logits: hid(32x64) @ gw2-tile(64x16)
        v8f ag[2] = { {}, {} };
        {
            const float* A0 = hidS + lm * HST + kg;
            const float* A1 = hidS + (16 + lm) * HST + kg;
            const v2f*   Bb = reinterpret_cast<const v2f*>(g2p + (size_t)(n0 >> 4) * 16 * 64) + lane;
            #pragma unroll
            for (int ks = 0; ks < 16; ++ks) {
                const v2f b  = Bb[ks * 32];
                v2f a0 = { A0[ks * 4], A0[ks * 4 + 1] };
                v2f a1 = { A1[ks * 4], A1[ks * 4 + 1] };
                ag[0] = wmma_f32(a0, b, ag[0]);
                ag[1] = wmma_f32(a1, b, ag[1]);
            }
        }
        // Laplacian diffusion: phi(32x1024) @ Lt-tile(1024x16)
        v8f ad[2] = { {}, {} };
        {
            const float* A0 = phiS + lm * PST + kg;
            const float* A1 = phiS + (16 + lm) * PST + kg;
            const v2f*   Bb = reinterpret_cast<const v2f*>(Lp + (size_t)(n0 >> 4) * 256 * 64) + lane;
            #pragma unroll 4
            for (int ks = 0; ks < 256; ++ks) {
                const v2f b  = Bb[ks * 32];
                v2f a0 = { A0[ks * 4], A0[ks * 4 + 1] };
                v2f a1 = { A1[ks * 4], A1[ks * 4 + 1] };
                ad[0] = wmma_f32(a0, b, ad[0]);
                ad[1] = wmma_f32(a1, b, ad[1]);
            }
        }

        const float pc  = prec[col];
        const float imc = im[col];
        const float gb  = gb2[col];
        #pragma unroll
        for (int rt = 0; rt < 2; ++rt) {
            #pragma unroll
            for (int r = 0; r < 8; ++r) {
                const int    m = rt * 16 + r + mhi;
                const size_t g = (size_t)(t0 + m) * HD + col;
                float pv  = phiS[m * PST + col];
                const float tgt = hT[g];
                float pd  = phidot[g];
                const float force = -(ad[rt][r] + (pv - tgt) * pc);
                const float gate  = 1.f / (1.f + __expf(-(ag[rt][r] + gb)));
                const float accel = (gate * force - BETA_ * pd) * imc;
                pd += DT_ * accel;
                pv += DT_ * pd;
                phidot[g] = pd;
                phi[g]    = pv;
            }
        }
    }
}

// ------------------------- layernorm ---------------------------------------
__global__ __launch_bounds__(256) void k_ln(const float* __restrict__ phi,
                                            const float* __restrict__ gamma,
                                            const float* __restrict__ beta,
                                            float* __restrict__ out) {
    const int tok = blockIdx.x;
    const int tid = threadIdx.x;
    const float* x = phi + (size_t)tok * HD;
    float s = 0.f, s2 = 0.f;
    for (int i = tid; i < HD; i += 256) {
        const float v = x[i];
        s += v; s2 += v * v;
    }
    for (int off = 16; off; off >>= 1) {
        s  += __shfl_down(s, off);
        s2 += __shfl_down(s2, off);
    }
    __shared__ float rs[8], rs2[8];
    if ((tid & 31) == 0) { rs[tid >> 5] = s; rs2[tid >> 5] = s2; }
    __syncthreads();
    __shared__ float mu_s, inv_s;
    if (tid == 0) {
        float a = 0.f, b = 0.f;
        for (int i = 0; i < 8; ++i) { a += rs[i]; b += rs2[i]; }
        const float mu = a / HD;
        mu_s  = mu;
        inv_s = rsqrtf(b / HD - mu * mu + 1e-5f);
    }
    __syncthreads();
    const float mu = mu_s, inv = inv_s;
    float* o = out + (size_t)tok * HD;
    for (int i = tid; i < HD; i += 256)
        o[i] = (x[i] - mu) * inv * gamma[i] + beta[i];
}

// ------------------------- launcher ----------------------------------------
extern "C" void kernel_launch(void* const* d_in, const int* in_sizes, int n_in,
                              void* d_out, int out_size, void* d_ws, size_t ws_size,
                              hipStream_t stream) {
    const float* h    = (const float*)d_in[0];
    const float* W    = (const float*)d_in[1];
    const float* mass = (const float*)d_in[2];
    const float* prec = (const float*)d_in[3];
    const float* gw1  = (const float*)d_in[4];
    const float* gb1  = (const float*)d_in[5];
    const float* gw2  = (const float*)d_in[6];
    const float* gb2  = (const float*)d_in[7];
    const float* lng  = (const float*)d_in[8];
    const float* lnb  = (const float*)d_in[9];
    float* out = (float*)d_out;

    float* ws   = (float*)d_ws;
    float* Lp   = ws;                                  // HD*HD   (packed B)
    float* g1p  = Lp  + (size_t)HD * HD;               // HD*BNK  (packed B)
    float* g2p  = g1p + (size_t)HD * BNK;              // BNK*HD  (packed B)
    float* deg  = g2p + (size_t)BNK * HD;              // HD
    float* im   = deg + HD;                            // HD
    float* phi  = im  + HD;                            // NTOK*HD
    float* pdot = phi + (size_t)NTOK * HD;             // NTOK*HD

    k_deg <<<HD,                 256, 0, stream>>>(W, mass, deg, im);
    k_ltp <<<(HD * HD) / 256,    256, 0, stream>>>(W, deg, Lp);
    k_gp  <<<(HD * BNK) / 256,   256, 0, stream>>>(gw1, gw2, g1p, g2p);
    k_init<<<(NTOK * HD) / 256,  256, 0, stream>>>(h, phi, pdot);

    for (int s = 0; s < NSET; ++s)
        k_step<<<NTOK / 32, 256, 0, stream>>>(h, Lp, g1p, gb1, g2p, gb2,
                                              prec, im, phi, pdot);

    k_ln<<<NTOK, 256, 0, stream>>>(phi, lng, lnb, out);
}